// RobustTopologyAwareGNN_12317966205311
// MI455X (gfx1250) — compile-verified
//
#include <hip/hip_runtime.h>
#include <hip/hip_bf16.h>

// ---------------------------------------------------------------------------
// RobustTopologyAwareGNN for MI455X (gfx1250, wave32).
// fp32 WMMA (v_wmma_f32_16x16x4_f32) for all dense GEMMs; the rest of the
// pipeline is L2-resident gather/scatter which dominates runtime.
// ---------------------------------------------------------------------------

typedef __attribute__((ext_vector_type(2))) float v2f;
typedef __attribute__((ext_vector_type(8))) float v8f;

#define GNN_N 40000
#define GNN_E 640000
#define GNN_F 64
#define GNN_H 128
#define GNN_L 3
#define GNN_EPS 1e-5f
#define GNN_SLOPE 0.1f

__device__ __forceinline__ float leaky(float x) {
    return x >= 0.0f ? x : GNN_SLOPE * x;
}

// ---------------------------------------------------------------------------
// init: deg = 1 (self loop), zero BN accumulators and pool accumulator
// ---------------------------------------------------------------------------
__global__ void init_kernel(float* __restrict__ deg, float* __restrict__ bnsum,
                            float* __restrict__ bnsq, float* __restrict__ pool,
                            int n) {
    int i = blockIdx.x * blockDim.x + threadIdx.x;
    if (i < n) deg[i] = 1.0f;
    if (i < GNN_F) { bnsum[i] = 0.0f; bnsq[i] = 0.0f; }
    if (i < GNN_H) pool[i] = 0.0f;
}

// ---------------------------------------------------------------------------
// in-degree count: deg[col[e]] += 1
// ---------------------------------------------------------------------------
__global__ void degree_kernel(const int* __restrict__ ei, float* __restrict__ deg,
                              int ne) {
    int e = blockIdx.x * blockDim.x + threadIdx.x;
    if (e < ne) atomicAdd(deg + ei[ne + e], 1.0f);
}

// ---------------------------------------------------------------------------
// BatchNorm statistics over the node dimension: 64 threads = 64 features,
// each block reduces 512 rows locally then atomically merges.
// ---------------------------------------------------------------------------
__global__ void bnstat_kernel(const float* __restrict__ x, float* __restrict__ bnsum,
                              float* __restrict__ bnsq, int nrows) {
    const int t = threadIdx.x;                 // feature 0..63
    const int r0 = blockIdx.x * 512;
    const int r1 = (r0 + 512 < nrows) ? (r0 + 512) : nrows;
    float s = 0.0f, q = 0.0f;
    for (int r = r0; r < r1; ++r) {
        float v = x[(size_t)r * GNN_F + t];
        s += v; q += v * v;
    }
    atomicAdd(bnsum + t, s);
    atomicAdd(bnsq + t, q);
}

// ---------------------------------------------------------------------------
// finalize: dinv = rsqrt(deg)  (in place over deg buffer), plus BN scale/shift
// ---------------------------------------------------------------------------
__global__ void finalize_kernel(float* __restrict__ degdinv,
                                const float* __restrict__ bnsum,
                                const float* __restrict__ bnsq,
                                const float* __restrict__ gamma,
                                const float* __restrict__ beta,
                                float* __restrict__ scale,
                                float* __restrict__ shift,
                                int nrows, float invN) {
    int i = blockIdx.x * blockDim.x + threadIdx.x;
    if (i < nrows) degdinv[i] = rsqrtf(degdinv[i]);
    if (i < GNN_F) {
        float mu  = bnsum[i] * invN;
        float var = bnsq[i] * invN - mu * mu;
        float sc  = gamma[i] * rsqrtf(var + GNN_EPS);
        scale[i]  = sc;
        shift[i]  = beta[i] - mu * sc;
    }
}

// ---------------------------------------------------------------------------
// apply BN elementwise: xn = x*scale[c] + shift[c]
// ---------------------------------------------------------------------------
__global__ void bnapply_kernel(const float* __restrict__ x,
                               const float* __restrict__ scale,
                               const float* __restrict__ shift,
                               float* __restrict__ xn, int total) {
    int i = blockIdx.x * blockDim.x + threadIdx.x;
    if (i < total) {
        int c = i & (GNN_F - 1);
        xn[i] = x[i] * scale[c] + shift[c];
    }
}

// ---------------------------------------------------------------------------
// fp32 WMMA GEMM: C[N,128] = A[N,K] @ W[K,128] (+bias).
// 8 waves per block; wave w owns rows [blk*128 + 16w, +16) and all 8 column
// tiles. W is staged transposed into dynamic LDS with padded stride (K+2) so
// each B fragment is one 8-byte ds load with spread banks. A fragments are
// 8-byte global loads (16-row working set stays in L0). N % 16 == 0, so
// every live wave has full EXEC for the WMMAs.
// ---------------------------------------------------------------------------
__global__ void __launch_bounds__(256)
gemm128_kernel(const float* __restrict__ A, const float* __restrict__ W,
               const float* __restrict__ bias, float* __restrict__ C,
               int nrows, int K) {
    extern __shared__ float smemW[];           // [128][K+2], Wt[c][k]
    const int kstride = K + 2;
    const int tid = threadIdx.x;

    // stage W transposed: W[k*128 + c] -> smemW[c*kstride + k]
    for (int idx = tid; idx < 128 * K; idx += 256) {
        int k = idx >> 7;
        int c = idx & 127;
        smemW[c * kstride + k] = W[idx];
    }
    __syncthreads();

    const int wave = tid >> 5;
    const int lane = tid & 31;
    const int row0 = blockIdx.x * 128 + wave * 16;
    if (row0 >= nrows) return;                 // wave-uniform predicate

    const int l15 = lane & 15;
    const int hi  = lane >> 4;                 // 0: K=k,k+1   1: K=k+2,k+3

    v8f acc[8] = {};

    const float* Arow = A + (size_t)(row0 + l15) * K + 2 * hi;
    for (int kk = 0; kk < K; kk += 4) {
        v2f a = *reinterpret_cast<const v2f*>(Arow + kk);
        #pragma unroll
        for (int n = 0; n < 8; ++n) {
            v2f b = *reinterpret_cast<const v2f*>(
                &smemW[(n * 16 + l15) * kstride + kk + 2 * hi]);
            acc[n] = __builtin_amdgcn_wmma_f32_16x16x4_f32(
                false, a, false, b, (short)0, acc[n], false, false);
        }
    }

    // C/D layout: VGPR v, lane -> row = row0 + v + 8*hi, col = n*16 + (lane&15)
    #pragma unroll
    for (int n = 0; n < 8; ++n) {
        const int col = n * 16 + l15;
        const float badd = bias ? bias[col] : 0.0f;
        #pragma unroll
        for (int v = 0; v < 8; ++v) {
            const int row = row0 + v + 8 * hi;
            C[(size_t)row * GNN_H + col] = acc[n][v] + badd;
        }
    }
}

// ---------------------------------------------------------------------------
// agg = hw * dinv^2 (self loop) + conv_b
// ---------------------------------------------------------------------------
__global__ void agginit_kernel(const float* __restrict__ hw,
                               const float* __restrict__ dinv,
                               const float* __restrict__ cb,
                               float* __restrict__ agg, int total) {
    int i = blockIdx.x * blockDim.x + threadIdx.x;
    if (i < total) {
        int n = i >> 7;
        int c = i & (GNN_H - 1);
        float d = dinv[n];
        agg[i] = hw[i] * d * d + cb[c];
    }
}

// ---------------------------------------------------------------------------
// edge scatter: one wave per edge; lane l handles features [4l, 4l+4).
// agg[col] += hw[row] * dinv[row]*dinv[col]   (f32 atomics, L2-resident)
// ---------------------------------------------------------------------------
__global__ void scatter_kernel(const float* __restrict__ hw,
                               const int* __restrict__ ei,
                               const float* __restrict__ dinv,
                               float* __restrict__ agg, int ne) {
    const int wave = threadIdx.x >> 5;
    const int lane = threadIdx.x & 31;
    const int e = blockIdx.x * 8 + wave;
    if (e >= ne) return;
    const int r = ei[e];
    const int c = ei[ne + e];
    const float s = dinv[r] * dinv[c];
    const float4 m = reinterpret_cast<const float4*>(hw + (size_t)r * GNN_H)[lane];
    float* dst = agg + (size_t)c * GNN_H + lane * 4;
    atomicAdd(dst + 0, m.x * s);
    atomicAdd(dst + 1, m.y * s);
    atomicAdd(dst + 2, m.z * s);
    atomicAdd(dst + 3, m.w * s);
}

// ---------------------------------------------------------------------------
// LayerNorm over H=128 + leaky + residual: one wave per node, float4/lane,
// wave32 butterfly reduction.
// ---------------------------------------------------------------------------
__global__ void ln_res_kernel(const float* __restrict__ agg,
                              const float* __restrict__ g,
                              const float* __restrict__ b,
                              float* __restrict__ h, int nrows) {
    const int wave = threadIdx.x >> 5;
    const int lane = threadIdx.x & 31;
    const int n = blockIdx.x * 8 + wave;
    if (n >= nrows) return;
    float4 v = reinterpret_cast<const float4*>(agg + (size_t)n * GNN_H)[lane];
    float s  = v.x + v.y + v.z + v.w;
    float sq = v.x * v.x + v.y * v.y + v.z * v.z + v.w * v.w;
    #pragma unroll
    for (int m = 16; m > 0; m >>= 1) {
        s  += __shfl_xor(s,  m, 32);
        sq += __shfl_xor(sq, m, 32);
    }
    const float mu  = s * (1.0f / GNN_H);
    const float var = sq * (1.0f / GNN_H) - mu * mu;
    const float rs  = rsqrtf(var + GNN_EPS);
    float4 ho = reinterpret_cast<const float4*>(h + (size_t)n * GNN_H)[lane];
    float4 gg = reinterpret_cast<const float4*>(g)[lane];
    float4 bb = reinterpret_cast<const float4*>(b)[lane];
    float4 o;
    o.x = leaky((v.x - mu) * rs * gg.x + bb.x) + ho.x;
    o.y = leaky((v.y - mu) * rs * gg.y + bb.y) + ho.y;
    o.z = leaky((v.z - mu) * rs * gg.z + bb.z) + ho.z;
    o.w = leaky((v.w - mu) * rs * gg.w + bb.w) + ho.w;
    reinterpret_cast<float4*>(h + (size_t)n * GNN_H)[lane] = o;
}

// ---------------------------------------------------------------------------
// global mean pool partials: 128 threads/block, thread t sums feature t over a
// grid-strided set of rows, then atomically merges.
// ---------------------------------------------------------------------------
__global__ void pool_kernel(const float* __restrict__ h, float* __restrict__ pool,
                            int nrows) {
    const int t = threadIdx.x;
    float s = 0.0f;
    for (int r = blockIdx.x; r < nrows; r += gridDim.x)
        s += h[(size_t)r * GNN_H + t];
    atomicAdd(pool + t, s);
}

// ---------------------------------------------------------------------------
// MLP head (single block, 128 threads): g = pool/N; leaky(LN(g@W1+b1)); @W2+b2
// ---------------------------------------------------------------------------
__global__ void mlp_kernel(const float* __restrict__ pool,
                           const float* __restrict__ W1, const float* __restrict__ b1,
                           const float* __restrict__ fg, const float* __restrict__ fb,
                           const float* __restrict__ W2, const float* __restrict__ b2,
                           float* __restrict__ out, float invN) {
    __shared__ float gv[GNN_H];
    __shared__ float z[GNN_H];
    __shared__ float red[GNN_H];
    const int t = threadIdx.x;
    gv[t] = pool[t] * invN;
    __syncthreads();
    float y = b1[t];
    for (int k = 0; k < GNN_H; ++k) y += gv[k] * W1[k * GNN_H + t];
    red[t] = y;
    __syncthreads();
    for (int s2 = 64; s2 > 0; s2 >>= 1) {
        if (t < s2) red[t] += red[t + s2];
        __syncthreads();
    }
    const float mu = red[0] * (1.0f / GNN_H);
    __syncthreads();
    const float d = y - mu;
    red[t] = d * d;
    __syncthreads();
    for (int s2 = 64; s2 > 0; s2 >>= 1) {
        if (t < s2) red[t] += red[t + s2];
        __syncthreads();
    }
    const float var = red[0] * (1.0f / GNN_H);
    const float rs  = rsqrtf(var + GNN_EPS);
    z[t] = leaky(d * rs * fg[t] + fb[t]);
    __syncthreads();
    float o = b2[t];
    for (int k = 0; k < GNN_H; ++k) o += z[k] * W2[k * GNN_H + t];
    out[t] = o;
}

// ---------------------------------------------------------------------------
extern "C" void kernel_launch(void* const* d_in, const int* in_sizes, int n_in,
                              void* d_out, int out_size, void* d_ws, size_t ws_size,
                              hipStream_t stream) {
    const float* x        = (const float*)d_in[0];
    const int*   ei       = (const int*)d_in[1];
    const float* bn_gamma = (const float*)d_in[2];
    const float* bn_beta  = (const float*)d_in[3];
    const float* W_emb    = (const float*)d_in[4];
    const float* b_emb    = (const float*)d_in[5];
    const float* conv_W   = (const float*)d_in[6];
    const float* conv_b   = (const float*)d_in[7];
    const float* ln_gamma = (const float*)d_in[8];
    const float* ln_beta  = (const float*)d_in[9];
    const float* W1       = (const float*)d_in[10];
    const float* b1       = (const float*)d_in[11];
    const float* fg       = (const float*)d_in[12];
    const float* fb       = (const float*)d_in[13];
    const float* W2       = (const float*)d_in[14];
    const float* b2       = (const float*)d_in[15];
    float* out = (float*)d_out;
    float* ws  = (float*)d_ws;

    const int N = GNN_N, E = GNN_E, F = GNN_F, H = GNN_H;

    // workspace layout (floats)
    float* deg   = ws;                      // [N]   -> becomes dinv in place
    float* bnsum = ws + N;                  // [64]
    float* bnsq  = bnsum + F;               // [64]
    float* scale = bnsq + F;                // [64]
    float* shift = scale + F;               // [64]
    float* pool  = shift + F;               // [128]
    float* xn    = pool + H;                // [N,64]
    float* h     = xn + (size_t)N * F;      // [N,128]
    float* hw    = h + (size_t)N * H;       // [N,128]
    float* agg   = hw + (size_t)N * H;      // [N,128]

    init_kernel<<<(N + 255) / 256, 256, 0, stream>>>(deg, bnsum, bnsq, pool, N);
    degree_kernel<<<(E + 255) / 256, 256, 0, stream>>>(ei, deg, E);
    bnstat_kernel<<<(N + 511) / 512, 64, 0, stream>>>(x, bnsum, bnsq, N);
    finalize_kernel<<<(N + 255) / 256, 256, 0, stream>>>(
        deg, bnsum, bnsq, bn_gamma, bn_beta, scale, shift, N, 1.0f / (float)N);
    bnapply_kernel<<<(N * F + 255) / 256, 256, 0, stream>>>(x, scale, shift, xn, N * F);

    // embed GEMM: h = xn @ W_emb + b_emb   (K=64)
    gemm128_kernel<<<(N + 127) / 128, 256, (size_t)128 * (F + 2) * 4, stream>>>(
        xn, W_emb, b_emb, h, N, F);

    for (int i = 0; i < GNN_L; ++i) {
        gemm128_kernel<<<(N + 127) / 128, 256, (size_t)128 * (H + 2) * 4, stream>>>(
            h, conv_W + (size_t)i * H * H, nullptr, hw, N, H);
        agginit_kernel<<<(N * H + 255) / 256, 256, 0, stream>>>(
            hw, deg, conv_b + (size_t)i * H, agg, N * H);
        scatter_kernel<<<E / 8, 256, 0, stream>>>(hw, ei, deg, agg, E);
        ln_res_kernel<<<(N + 7) / 8, 256, 0, stream>>>(
            agg, ln_gamma + (size_t)i * H, ln_beta + (size_t)i * H, h, N);
    }

    pool_kernel<<<256, 128, 0, stream>>>(h, pool, N);
    mlp_kernel<<<1, 128, 0, stream>>>(pool, W1, b1, fg, fb, W2, b2, out,
                                      1.0f / (float)N);
}